// ComplexityAnalyzer_90580860273225
// MI455X (gfx1250) — compile-verified
//
#include <hip/hip_runtime.h>

typedef __attribute__((ext_vector_type(2))) float v2f;
typedef __attribute__((ext_vector_type(8))) float v8f;

#define HD 64   // hidden dim
#define FD 10   // feature dim
#define OD 5    // output dim

// ---------------------------------------------------------------------------
// h = relu(features @ W_node^T + b_node)   (K=10, scalar VALU is fine)
// ---------------------------------------------------------------------------
__global__ __launch_bounds__(256)
void node_linear_kernel(const float* __restrict__ feat,
                        const float* __restrict__ Wn,
                        const float* __restrict__ bn,
                        float* __restrict__ h, long long total)
{
    long long t = (long long)blockIdx.x * blockDim.x + threadIdx.x;
    if (t >= total) return;
    int o = (int)(t & (HD - 1));
    long long i = t >> 6;
    const float* f = feat + i * FD;
    const float* w = Wn + (long long)o * FD;
    float acc = bn[o];
#pragma unroll
    for (int k = 0; k < FD; ++k) acc += f[k] * w[k];
    h[t] = acc > 0.f ? acc : 0.f;
}

// ---------------------------------------------------------------------------
// zero fill (float4 granularity)
// ---------------------------------------------------------------------------
__global__ __launch_bounds__(256)
void fill_zero_kernel(float4* __restrict__ p, long long n4)
{
    long long t = (long long)blockIdx.x * blockDim.x + threadIdx.x;
    if (t < n4) p[t] = make_float4(0.f, 0.f, 0.f, 0.f);
}

// ---------------------------------------------------------------------------
// agg[dst] += h[src] for every edge.  16 threads per edge, each handles a
// float4 chunk of the 64-wide row: 1 b128 load + 4 global_atomic_add_f32.
// Rows live in the 192MB L2, so the random traffic stays on-chip.
// ---------------------------------------------------------------------------
__global__ __launch_bounds__(256)
void scatter_kernel(const float* __restrict__ h,
                    const int* __restrict__ edges,
                    float* __restrict__ agg, long long nEdges)
{
    long long t = (long long)blockIdx.x * blockDim.x + threadIdx.x;
    if (t >= nEdges * 16) return;
    long long e = t >> 4;
    int chunk = (int)(t & 15);
    long long src = (long long)edges[e * 2 + 0];
    long long dst = (long long)edges[e * 2 + 1];
    const float4 v = *(const float4*)(h + src * HD + chunk * 4);
    float* d = agg + dst * HD + chunk * 4;
    atomicAdd(d + 0, v.x);
    atomicAdd(d + 1, v.y);
    atomicAdd(d + 2, v.z);
    atomicAdd(d + 3, v.w);
}

// ---------------------------------------------------------------------------
// out = relu((h + agg) @ Wc^T + bc) using V_WMMA_F32_16X16X4_F32 (fp32 exact).
// One wave -> one 16x16 output tile, 16 WMMA steps over K=64.
// Block = 128 threads = 4 waves covering the 4 N-tiles of 16 rows.
//
// A-frag (16x4 f32, 2 VGPRs): lane L holds X[L%16][k0 + 2*(L/16) + {0,1}]
// B-frag (4x16 f32, 2 VGPRs): lane L holds W[n0+L%16][k0 + 2*(L/16) + {0,1}]
// C/D (16x16 f32, 8 VGPRs):   vgpr r, lane L -> row r + 8*(L/16), col L%16
// ---------------------------------------------------------------------------
__global__ __launch_bounds__(128)
void conv_wmma_kernel(const float* __restrict__ h,
                      const float* __restrict__ agg,
                      const float* __restrict__ W,
                      const float* __restrict__ bias,
                      float* __restrict__ out)
{
    const int lane = threadIdx.x & 31;
    const int wv   = threadIdx.x >> 5;   // which 16-wide N tile (0..3)
    const int m    = lane & 15;
    const int half = lane >> 4;
    const int n0   = wv * 16;
    const int nc   = n0 + m;             // column this lane serves for B/bias

    const long long row = (long long)blockIdx.x * 16 + m;
    const float* __restrict__ hrow = h   + row * HD;
    const float* __restrict__ arow = agg + row * HD;
    const float* __restrict__ wrow = W   + (long long)nc * HD;

    v8f c = {};
#pragma unroll
    for (int k0 = 0; k0 < HD; k0 += 4) {
        const int k = k0 + 2 * half;
        v2f a, b;
        a.x = hrow[k]     + arow[k];
        a.y = hrow[k + 1] + arow[k + 1];
        b.x = wrow[k];
        b.y = wrow[k + 1];
        c = __builtin_amdgcn_wmma_f32_16x16x4_f32(
                /*neg_a=*/false, a, /*neg_b=*/false, b,
                /*c_mod=*/(short)0, c, /*reuse_a=*/false, /*reuse_b=*/false);
    }

    const float bb = bias[nc];
    const long long rbase = (long long)blockIdx.x * 16 + 8 * half;
#pragma unroll
    for (int r = 0; r < 8; ++r) {
        float y = c[r] + bb;
        out[(rbase + r) * HD + n0 + m] = y > 0.f ? y : 0.f;
    }
}

// scalar fallback for rows beyond the last full 16-row tile (not hit for N=200000)
__global__ __launch_bounds__(256)
void conv_tail_kernel(const float* __restrict__ h, const float* __restrict__ agg,
                      const float* __restrict__ W, const float* __restrict__ bias,
                      float* __restrict__ out, int rowStart, int n)
{
    int t = blockIdx.x * blockDim.x + threadIdx.x;
    int rows = n - rowStart;
    if (t >= rows * HD) return;
    long long r = rowStart + (t >> 6);
    int o = t & (HD - 1);
    float acc = bias[o];
    for (int k = 0; k < HD; ++k)
        acc += (h[r * HD + k] + agg[r * HD + k]) * W[(long long)o * HD + k];
    out[r * HD + o] = acc > 0.f ? acc : 0.f;
}

// ---------------------------------------------------------------------------
// column sums of h (N x 64) -> gsum[64] via LDS partials + few atomics
// ---------------------------------------------------------------------------
__global__ __launch_bounds__(256)
void colsum_kernel(const float* __restrict__ h, float* __restrict__ gsum, int n)
{
    __shared__ float s[256];
    const int col  = threadIdx.x & (HD - 1);
    const int rgrp = threadIdx.x >> 6;   // 0..3
    float acc = 0.f;
    for (long long i = (long long)blockIdx.x * 4 + rgrp; i < n; i += (long long)gridDim.x * 4)
        acc += h[i * HD + col];
    s[rgrp * HD + col] = acc;
    __syncthreads();
    if (rgrp == 0) {
        float v = s[col] + s[HD + col] + s[2 * HD + col] + s[3 * HD + col];
        atomicAdd(&gsum[col], v);
    }
}

// ---------------------------------------------------------------------------
// metrics = (gsum / N) @ W_out^T + b_out   (5 outputs, one wave)
// ---------------------------------------------------------------------------
__global__ __launch_bounds__(32)
void final_kernel(const float* __restrict__ gsum,
                  const float* __restrict__ Wout,
                  const float* __restrict__ bout,
                  float* __restrict__ out, float invN)
{
    int o = threadIdx.x;
    if (o < OD) {
        float acc = bout[o];
#pragma unroll
        for (int k = 0; k < HD; ++k)
            acc += (gsum[k] * invN) * Wout[o * HD + k];
        out[o] = acc;
    }
}

// ---------------------------------------------------------------------------
extern "C" void kernel_launch(void* const* d_in, const int* in_sizes, int n_in,
                              void* d_out, int out_size, void* d_ws, size_t ws_size,
                              hipStream_t stream)
{
    // inputs (setup_inputs order): nodes, features, edges, W_node, b_node,
    //                              W_conv1, b_conv1, W_out, b_out
    const float* features = (const float*)d_in[1];
    const int*   edges    = (const int*)d_in[2];
    const float* W_node   = (const float*)d_in[3];
    const float* b_node   = (const float*)d_in[4];
    const float* W_conv   = (const float*)d_in[5];
    const float* b_conv   = (const float*)d_in[6];
    const float* W_out    = (const float*)d_in[7];
    const float* b_out    = (const float*)d_in[8];

    const int       n  = in_sizes[1] / FD;          // 200000
    const long long nE = (long long)in_sizes[2] / 2; // 3200000

    float* hA   = (float*)d_ws;
    float* hB   = hA + (size_t)n * HD;
    float* agg  = hB + (size_t)n * HD;
    float* gsum = agg + (size_t)n * HD;

    const long long totNH = (long long)n * HD;

    // 1. node linear
    node_linear_kernel<<<(unsigned)((totNH + 255) / 256), 256, 0, stream>>>(
        features, W_node, b_node, hA, totNH);

    // 2. two graph-conv layers (ping-pong hA <-> hB)
    float* cur = hA;
    float* nxt = hB;
    const int nTiles = n / 16;
    const int tailStart = nTiles * 16;
    for (int layer = 0; layer < 2; ++layer) {
        const long long n4 = totNH / 4;
        fill_zero_kernel<<<(unsigned)((n4 + 255) / 256), 256, 0, stream>>>(
            (float4*)agg, n4);
        const long long st = nE * 16;
        scatter_kernel<<<(unsigned)((st + 255) / 256), 256, 0, stream>>>(
            cur, edges, agg, nE);
        if (nTiles > 0)
            conv_wmma_kernel<<<(unsigned)nTiles, 128, 0, stream>>>(
                cur, agg, W_conv, b_conv, nxt);
        if (tailStart < n) {
            int tailElems = (n - tailStart) * HD;
            conv_tail_kernel<<<(tailElems + 255) / 256, 256, 0, stream>>>(
                cur, agg, W_conv, b_conv, nxt, tailStart, n);
        }
        float* tmp = cur; cur = nxt; nxt = tmp;
    }

    // 3. mean over nodes + final projection
    fill_zero_kernel<<<1, 256, 0, stream>>>((float4*)gsum, HD / 4);
    colsum_kernel<<<256, 256, 0, stream>>>(cur, gsum, n);
    final_kernel<<<1, 32, 0, stream>>>(gsum, W_out, b_out, (float*)d_out,
                                       1.0f / (float)n);
}